// graphnas_module_20383914787269
// MI455X (gfx1250) — compile-verified
//
#include <hip/hip_runtime.h>
#include <math.h>

#define N_NODES 40000
#define N_EDGES 320000
#define ETOT    (N_EDGES + N_NODES)   // 360000 after self loops
#define F_IN    256
#define H1      4
#define C1      128
#define D1      (H1*C1)               // 512
#define H2      8
#define C2      6
#define D2      (H2*C2)               // 48
#define BN_EPS    1e-5f
#define NEG_SLOPE 0.2f

typedef __attribute__((ext_vector_type(2))) float v2f;
typedef __attribute__((ext_vector_type(8))) float v8f;

// ---------------- helpers ----------------

__device__ __forceinline__ void atomic_max_f32(float* addr, float v) {
  // IEEE bit-trick: int-max for non-negative, uint-min for negative.
  if (v >= 0.0f) atomicMax((int*)addr, __float_as_int(v));
  else           atomicMin((unsigned int*)addr, __float_as_uint(v));
}

__device__ __forceinline__ void edge_rc(const int* __restrict__ rowA,
                                        const int* __restrict__ colA,
                                        int e, int& r, int& c) {
  if (e < N_EDGES) { r = rowA[e]; c = colA[e]; }
  else             { r = e - N_EDGES; c = r; }   // appended self loops
}

// ---------------- init ----------------

__global__ void fill_f32(float* __restrict__ p, float v, int n) {
  int i = blockIdx.x * blockDim.x + threadIdx.x;
  int stride = gridDim.x * blockDim.x;
  for (; i < n; i += stride) p[i] = v;
}

// bn scale/shift; b1 folded into layer-2 shift since reference adds bias before BN2.
__global__ void bn_prep(const float* g1, const float* be1, const float* mu1, const float* va1,
                        const float* g2, const float* be2, const float* mu2, const float* va2,
                        const float* bias1,
                        float* sc1, float* sh1, float* sc2, float* sh2) {
  int i = blockIdx.x * blockDim.x + threadIdx.x;
  if (i < F_IN) {
    float s = g1[i] / sqrtf(va1[i] + BN_EPS);
    sc1[i] = s; sh1[i] = be1[i] - mu1[i] * s;
  }
  if (i < D1) {
    float s = g2[i] / sqrtf(va2[i] + BN_EPS);
    sc2[i] = s; sh2[i] = be2[i] + (bias1[i] - mu2[i]) * s;
  }
}

// ---------------- f32 WMMA GEMM with fused per-column BN on A ----------------
// C[M,N] = (A[M,K]*scale[k]+shift[k]) @ B[K,N].  One 16x16 tile per wave.
// M,N multiples of 16 (40000/16=2500 exact).

__global__ __launch_bounds__(256)
void gemm_bn_wmma(const float* __restrict__ A,
                  const float* __restrict__ scale,
                  const float* __restrict__ shift,
                  const float* __restrict__ B,
                  float* __restrict__ C,
                  int M, int K, int N) {
  int wave = blockIdx.x * (blockDim.x >> 5) + (threadIdx.x >> 5);
  int ntiles = N >> 4;
  int tm = wave / ntiles;
  int tn = wave - tm * ntiles;
  if (tm >= (M >> 4)) return;                 // wave-uniform: EXEC stays all-ones

  int lane = threadIdx.x & 31;
  int half = lane >> 4;                        // 0: K pair {0,1}; 1: K pair {2,3}
  int l16  = lane & 15;
  int row  = (tm << 4) + l16;                  // A: M = lane%16 in both halves
  int colB = (tn << 4) + l16;                  // B/C: N = lane%16

  const float* __restrict__ Arow = A + (size_t)row * K;
  v8f acc = {};
  for (int k = 0; k < K; k += 4) {
    int kk = k + (half << 1);
    v2f a, b;
    a.x = Arow[kk]     * scale[kk]     + shift[kk];
    a.y = Arow[kk + 1] * scale[kk + 1] + shift[kk + 1];
    b.x = B[(size_t)kk * N + colB];
    b.y = B[(size_t)(kk + 1) * N + colB];
    acc = __builtin_amdgcn_wmma_f32_16x16x4_f32(
        /*neg_a=*/false, a, /*neg_b=*/false, b,
        /*c_mod=*/(short)0, acc, /*reuse_a=*/false, /*reuse_b=*/false);
  }
  // C/D layout: VGPR r -> M = r + 8*half, N = lane%16
  float* Crow = C + (size_t)((tm << 4) + (half << 3)) * N + colB;
  #pragma unroll
  for (int r = 0; r < 8; ++r) Crow[(size_t)r * N] = acc[r];
}

// ---------------- layer 1 (GAT) attention ----------------

// ai[n,h] = sum_c h1[n,h,c]*att_l[h,c] ; aj likewise with att_r. One wave per (n,h).
__global__ void node_att1(const float* __restrict__ h1, const float* __restrict__ att1,
                          float* __restrict__ ai, float* __restrict__ aj) {
  int wave = blockIdx.x * (blockDim.x >> 5) + (threadIdx.x >> 5);
  if (wave >= N_NODES * H1) return;
  int n = wave >> 2, h = wave & 3;
  int lane = threadIdx.x & 31;
  const float* __restrict__ hp = h1 + (size_t)n * D1 + h * C1;
  const float* __restrict__ al = att1 + h * (2 * C1);
  const float* __restrict__ ar = al + C1;
  float si = 0.f, sj = 0.f;
  #pragma unroll
  for (int c0 = 0; c0 < C1; c0 += 32) {
    float v = hp[c0 + lane];
    si += v * al[c0 + lane];
    sj += v * ar[c0 + lane];
  }
  #pragma unroll
  for (int off = 16; off > 0; off >>= 1) {
    si += __shfl_xor(si, off, 32);
    sj += __shfl_xor(sj, off, 32);
  }
  if (lane == 0) { ai[wave] = si; aj[wave] = sj; }
}

__global__ void edge_max1(const int* __restrict__ rowA, const int* __restrict__ colA,
                          const float* __restrict__ ai, const float* __restrict__ aj,
                          float* __restrict__ m1) {
  int e = blockIdx.x * blockDim.x + threadIdx.x;
  if (e >= ETOT) return;
  int row, col; edge_rc(rowA, colA, e, row, col);
  float4 av = *(const float4*)(ai + (size_t)row * 4);
  float4 bv = *(const float4*)(aj + (size_t)col * 4);
  float s[4] = {av.x + bv.x, av.y + bv.y, av.z + bv.z, av.w + bv.w};
  #pragma unroll
  for (int h = 0; h < 4; ++h) {
    float a = s[h]; a = a > 0.f ? a : NEG_SLOPE * a;
    atomic_max_f32(&m1[row * 4 + h], a);
  }
}

__global__ void edge_sum1(const int* __restrict__ rowA, const int* __restrict__ colA,
                          const float* __restrict__ ai, const float* __restrict__ aj,
                          const float* __restrict__ m1, float* __restrict__ s1) {
  int e = blockIdx.x * blockDim.x + threadIdx.x;
  if (e >= ETOT) return;
  int row, col; edge_rc(rowA, colA, e, row, col);
  float4 av = *(const float4*)(ai + (size_t)row * 4);
  float4 bv = *(const float4*)(aj + (size_t)col * 4);
  float4 mv = *(const float4*)(m1 + (size_t)row * 4);
  float s[4] = {av.x + bv.x, av.y + bv.y, av.z + bv.z, av.w + bv.w};
  float m[4] = {mv.x, mv.y, mv.z, mv.w};
  #pragma unroll
  for (int h = 0; h < 4; ++h) {
    float a = s[h]; a = a > 0.f ? a : NEG_SLOPE * a;
    atomicAdd(&s1[row * 4 + h], expf(a - m[h]));
  }
}

// One wave per (edge, head): gather 128 floats from h1[col], scatter-add to out1[row].
__global__ __launch_bounds__(256)
void edge_scatter1(const int* __restrict__ rowA, const int* __restrict__ colA,
                   const float* __restrict__ ai, const float* __restrict__ aj,
                   const float* __restrict__ m1, const float* __restrict__ s1,
                   const float* __restrict__ h1, float* __restrict__ out1) {
  int wave = blockIdx.x * (blockDim.x >> 5) + (threadIdx.x >> 5);
  if (wave >= ETOT * H1) return;
  int e = wave >> 2, h = wave & 3;
  int row, col; edge_rc(rowA, colA, e, row, col);
  float a = ai[row * 4 + h] + aj[col * 4 + h];
  a = a > 0.f ? a : NEG_SLOPE * a;
  float w = expf(a - m1[row * 4 + h]) / (s1[row * 4 + h] + 1e-16f);
  int lane = threadIdx.x & 31;
  const float4* __restrict__ src = (const float4*)(h1 + (size_t)col * D1 + h * C1);
  float4 v = src[lane];
  float* dst = out1 + (size_t)row * D1 + h * C1 + lane * 4;
  atomicAdd(dst + 0, v.x * w);
  atomicAdd(dst + 1, v.y * w);
  atomicAdd(dst + 2, v.z * w);
  atomicAdd(dst + 3, v.w * w);
}

// ---------------- layer 2 ('linear' attention) ----------------

// alpha depends only on source node: s2[n,h] = tanh(sum_c h2[n,h,c]*(att_l+att_r)[h,c])
__global__ void node_att2(const float* __restrict__ h2, const float* __restrict__ att2,
                          float* __restrict__ s2) {
  int tid = blockIdx.x * blockDim.x + threadIdx.x;
  if (tid >= N_NODES * H2) return;
  int n = tid >> 3, h = tid & 7;
  const float* __restrict__ hp = h2 + (size_t)n * D2 + h * C2;
  const float* __restrict__ ap = att2 + h * (2 * C2);
  float s = 0.f;
  #pragma unroll
  for (int c = 0; c < C2; ++c) s += hp[c] * (ap[c] + ap[C2 + c]);
  s2[tid] = tanhf(s);
}

__global__ void edge_max2(const int* __restrict__ rowA, const int* __restrict__ colA,
                          const float* __restrict__ s2, float* __restrict__ m2) {
  int e = blockIdx.x * blockDim.x + threadIdx.x;
  if (e >= ETOT) return;
  int row, col; edge_rc(rowA, colA, e, row, col);
  const float* sv = s2 + (size_t)col * 8;
  float* mr = m2 + (size_t)row * 8;
  #pragma unroll
  for (int h = 0; h < 8; ++h) atomic_max_f32(&mr[h], sv[h]);
}

__global__ void edge_sum2(const int* __restrict__ rowA, const int* __restrict__ colA,
                          const float* __restrict__ s2, const float* __restrict__ m2,
                          float* __restrict__ ss2) {
  int e = blockIdx.x * blockDim.x + threadIdx.x;
  if (e >= ETOT) return;
  int row, col; edge_rc(rowA, colA, e, row, col);
  const float* sv = s2 + (size_t)col * 8;
  const float* mr = m2 + (size_t)row * 8;
  #pragma unroll
  for (int h = 0; h < 8; ++h) atomicAdd(&ss2[row * 8 + h], expf(sv[h] - mr[h]));
}

// One thread per (edge, head): 6 atomic adds.
__global__ void edge_scatter2(const int* __restrict__ rowA, const int* __restrict__ colA,
                              const float* __restrict__ s2, const float* __restrict__ m2,
                              const float* __restrict__ ss2,
                              const float* __restrict__ h2, float* __restrict__ out2) {
  int tid = blockIdx.x * blockDim.x + threadIdx.x;
  if (tid >= ETOT * H2) return;
  int e = tid >> 3, h = tid & 7;
  int row, col; edge_rc(rowA, colA, e, row, col);
  float w = expf(s2[col * 8 + h] - m2[row * 8 + h]) / (ss2[row * 8 + h] + 1e-16f);
  const float* __restrict__ src = h2 + (size_t)col * D2 + h * C2;
  float* dst = out2 + (size_t)row * D2 + h * C2;
  #pragma unroll
  for (int c = 0; c < C2; ++c) atomicAdd(&dst[c], src[c] * w);
}

// mean over heads + bias + elu + log_softmax
__global__ void finalize(const float* __restrict__ out2, const float* __restrict__ b2,
                         float* __restrict__ out) {
  int n = blockIdx.x * blockDim.x + threadIdx.x;
  if (n >= N_NODES) return;
  float v[C2];
  #pragma unroll
  for (int c = 0; c < C2; ++c) {
    float s = 0.f;
    #pragma unroll
    for (int h = 0; h < H2; ++h) s += out2[(size_t)n * D2 + h * C2 + c];
    float t = s * (1.0f / H2) + b2[c];
    v[c] = t > 0.f ? t : (expf(t) - 1.0f);     // elu
  }
  float m = v[0];
  #pragma unroll
  for (int c = 1; c < C2; ++c) m = fmaxf(m, v[c]);
  float sum = 0.f;
  #pragma unroll
  for (int c = 0; c < C2; ++c) sum += expf(v[c] - m);
  float lse = logf(sum);
  #pragma unroll
  for (int c = 0; c < C2; ++c) out[(size_t)n * C2 + c] = v[c] - m - lse;
}

// ---------------- launcher ----------------

extern "C" void kernel_launch(void* const* d_in, const int* in_sizes, int n_in,
                              void* d_out, int out_size, void* d_ws, size_t ws_size,
                              hipStream_t stream) {
  const float* x    = (const float*)d_in[0];
  const int*   ei   = (const int*)  d_in[1];
  const float* g1   = (const float*)d_in[2];
  const float* be1  = (const float*)d_in[3];
  const float* mu1  = (const float*)d_in[4];
  const float* va1  = (const float*)d_in[5];
  const float* W1   = (const float*)d_in[6];
  const float* att1 = (const float*)d_in[7];
  const float* b1   = (const float*)d_in[8];
  const float* g2   = (const float*)d_in[9];
  const float* be2  = (const float*)d_in[10];
  const float* mu2  = (const float*)d_in[11];
  const float* va2  = (const float*)d_in[12];
  const float* W2   = (const float*)d_in[13];
  const float* att2 = (const float*)d_in[14];
  const float* b2   = (const float*)d_in[15];
  const int* rowA = ei;
  const int* colA = ei + N_EDGES;
  float* out = (float*)d_out;
  float* ws  = (float*)d_ws;

  // workspace layout (floats); all offsets 16B-aligned
  float* sc1  = ws;                                 // 256
  float* sh1  = ws + 256;                           // 256
  float* sc2  = ws + 512;                           // 512
  float* sh2  = ws + 1024;                          // 512
  float* ai   = ws + 1536;                          // N*4
  float* aj   = ai  + (size_t)N_NODES * H1;         // N*4
  float* m1   = aj  + (size_t)N_NODES * H1;         // N*4
  float* s1   = m1  + (size_t)N_NODES * H1;         // N*4
  float* out1 = s1  + (size_t)N_NODES * H1;         // N*512 (layer1 aggregate)
  float* h1   = out1 + (size_t)N_NODES * D1;        // N*512 (dead after scatter1)
  // layer-2 arrays overlay the dead h1 region
  float* h2   = h1;                                 // N*48
  float* s2   = h2  + (size_t)N_NODES * D2;         // N*8
  float* m2   = s2  + (size_t)N_NODES * H2;         // N*8
  float* ss2  = m2  + (size_t)N_NODES * H2;         // N*8
  float* out2 = ss2 + (size_t)N_NODES * H2;         // N*48

  const float NEG_INF = -__builtin_inff();

  bn_prep<<<2, 256, 0, stream>>>(g1, be1, mu1, va1, g2, be2, mu2, va2, b1,
                                 sc1, sh1, sc2, sh2);
  fill_f32<<<625,   256, 0, stream>>>(m1, NEG_INF, N_NODES * H1);
  fill_f32<<<625,   256, 0, stream>>>(s1, 0.0f,    N_NODES * H1);
  fill_f32<<<20000, 256, 0, stream>>>(out1, 0.0f,  N_NODES * D1);

  // GEMM1: h1 = bn1(x) @ W1   (2500*32 tiles, 8 waves/block)
  gemm_bn_wmma<<<10000, 256, 0, stream>>>(x, sc1, sh1, W1, h1, N_NODES, F_IN, D1);

  node_att1<<<20000, 256, 0, stream>>>(h1, att1, ai, aj);
  edge_max1<<<(ETOT + 255) / 256, 256, 0, stream>>>(rowA, colA, ai, aj, m1);
  edge_sum1<<<(ETOT + 255) / 256, 256, 0, stream>>>(rowA, colA, ai, aj, m1, s1);
  edge_scatter1<<<(ETOT * H1) / 8, 256, 0, stream>>>(rowA, colA, ai, aj, m1, s1, h1, out1);

  // layer-2 inits (after scatter1: overlays the h1 region)
  fill_f32<<<1250, 256, 0, stream>>>(m2, NEG_INF, N_NODES * H2);
  fill_f32<<<1250, 256, 0, stream>>>(ss2, 0.0f,   N_NODES * H2);
  fill_f32<<<7500, 256, 0, stream>>>(out2, 0.0f,  N_NODES * D2);

  // GEMM2: h2 = bn2(out1 + b1) @ W2   (2500*3 tiles)
  gemm_bn_wmma<<<(2500 * 3 + 7) / 8, 256, 0, stream>>>(out1, sc2, sh2, W2, h2,
                                                       N_NODES, D1, D2);

  node_att2<<<(N_NODES * H2) / 256, 256, 0, stream>>>(h2, att2, s2);
  edge_max2<<<(ETOT + 255) / 256, 256, 0, stream>>>(rowA, colA, s2, m2);
  edge_sum2<<<(ETOT + 255) / 256, 256, 0, stream>>>(rowA, colA, s2, m2, ss2);
  edge_scatter2<<<(ETOT * H2) / 256, 256, 0, stream>>>(rowA, colA, s2, m2, ss2, h2, out2);

  finalize<<<(N_NODES + 255) / 256, 256, 0, stream>>>(out2, b2, out);
}